// AdditiveAttention_5798205849844
// MI455X (gfx1250) — compile-verified
//
#include <hip/hip_runtime.h>
#include <hip/hip_bf16.h>

// ---------------- types ----------------
typedef float v2f __attribute__((ext_vector_type(2)));
typedef float v8f __attribute__((ext_vector_type(8)));

#define D_DIM   256
#define LKV     512
#define LQ      256
#define BATCH   4
#define NEG_INF (-__builtin_inff())

// scoring kernel tiling
#define Q_TILE  4      // queries per block (kp reuse factor)
#define K_BLK   256    // k-range per block (one thread per k)
#define D_CHUNK 32     // d staged per chunk
#define KT_PITCH 36    // LDS row pitch (multiple of 4 for b128; 2-way conflict worst case)

// ---------------- fast tanh: v_tanh_f32 if available ----------------
__device__ __forceinline__ float fast_tanh(float x) {
#if __has_builtin(__builtin_amdgcn_tanhf)
  return __builtin_amdgcn_tanhf(x);
#else
  float e = __expf(2.0f * x);
  return 1.0f - 2.0f / (e + 1.0f);
#endif
}

__device__ __forceinline__ float fget(const float4& v, int u) {
  const float* p = &v.x;
  return p[u];   // folds to component select under full unroll
}

// =====================================================================
// Kernel 1: projection GEMM via V_WMMA_F32_16X16X4_F32
//   Out[m, n] = sum_k X[m,k] * W[n,k]  (+ bias[n])
//   One wave computes one 16x16 tile; 64 chained WMMAs over K=256.
// =====================================================================
__global__ __launch_bounds__(256) void proj_wmma_kernel(
    const float* __restrict__ X,
    const float* __restrict__ W,     // already offset for Wq / Wk; row stride 512
    const float* __restrict__ bias,  // per-n, may be null
    float* __restrict__ Out,         // M x 256
    int M)
{
  const int lane = threadIdx.x & 31;
  const int wave = threadIdx.x >> 5;
  const int tile = blockIdx.x * 8 + wave;   // one 16x16 tile per wave
  const int tm   = tile >> 4;               // 256/16 = 16 n-tiles
  const int tn   = tile & 15;
  if (tm * 16 >= M) return;                 // wave-uniform guard (EXEC all-1 at WMMA)

  const int lm   = lane & 15;
  const int half = lane >> 4;               // K sub-pair select per ISA layout

  const float* a_ptr = X + (size_t)(tm * 16 + lm) * D_DIM + half * 2;
  const float* b_ptr = W + (size_t)(tn * 16 + lm) * 512 + half * 2;

  v8f c = {};
#pragma unroll 8
  for (int k0 = 0; k0 < D_DIM; k0 += 4) {
    v2f a = *(const v2f*)(a_ptr + k0);
    v2f b = *(const v2f*)(b_ptr + k0);
    c = __builtin_amdgcn_wmma_f32_16x16x4_f32(false, a, false, b, (short)0, c,
                                              false, false);
  }

  const float bv = bias ? bias[tn * 16 + lm] : 0.0f;
  float* o = Out + (size_t)(tm * 16 + half * 8) * D_DIM + tn * 16 + lm;
#pragma unroll
  for (int r = 0; r < 8; ++r)
    o[(size_t)r * D_DIM] = c[r] + bv;
}

// =====================================================================
// Kernel 2: logits[b,q,k] = sum_d tanh(qp[b,q,d] + kp[b,k,d]) * w[d] + bl
//   Thread-per-k (no cross-lane reductions), Q_TILE=4 queries share each
//   staged kp element. kp staged through LDS in 32-d chunks, coalesced.
//   Writes raw logits into d_out; softmax pass follows.
//   Grid: BATCH * (LQ/Q_TILE) * (LKV/K_BLK) = 4*64*2 = 512 blocks x 256 thr.
// =====================================================================
__global__ __launch_bounds__(256) void score_kernel(
    const float* __restrict__ qp,      // (B*LQ) x 256
    const float* __restrict__ kp,      // (B*LKV) x 256
    const float* __restrict__ w_logit, // 256
    const float* __restrict__ b_logit, // 1
    float* __restrict__ out)           // B x LQ x LKV (raw logits)
{
  __shared__ float qsh4[D_DIM * Q_TILE];       // [d][qi] interleaved -> b128 broadcast
  __shared__ float wsh[D_DIM];
  __shared__ float kt[K_BLK * KT_PITCH];       // [local k][d-chunk], pitch 36

  const int tid = threadIdx.x;
  const int bid = blockIdx.x;
  const int ks  = bid & 1;                     // k half
  const int qt  = (bid >> 1) & 63;             // q tile
  const int b   = bid >> 7;                    // batch
  const int q0  = qt * Q_TILE;
  const int k0  = ks * K_BLK;
  const int k   = k0 + tid;                    // this thread's k (local row = tid)

  // stage q rows (interleaved) and w_logit
#pragma unroll
  for (int qi = 0; qi < Q_TILE; ++qi)
    qsh4[tid * Q_TILE + qi] = qp[(size_t)(b * LQ + q0 + qi) * D_DIM + tid];
  wsh[tid] = w_logit[tid];

  float acc[Q_TILE] = {0.f, 0.f, 0.f, 0.f};
  const float* kp_b = kp + (size_t)b * LKV * D_DIM;

  for (int d0 = 0; d0 < D_DIM; d0 += D_CHUNK) {
    __syncthreads();   // kt reuse guard (also covers initial qsh/wsh writes)
    // stage kt[row][0..31] = kp[b, k0+row, d0..d0+31]; 2048 float4s, coalesced
#pragma unroll
    for (int i = 0; i < 8; ++i) {
      const int idx = tid + i * 256;           // 0..2047
      const int row = idx >> 3;
      const int j4  = idx & 7;
      const float4 v = *(const float4*)(kp_b + (size_t)(k0 + row) * D_DIM + d0 + j4 * 4);
      *(float4*)&kt[row * KT_PITCH + j4 * 4] = v;   // conflict-free (rows in wave differ <4)
    }
    __syncthreads();

    const float* ktr = &kt[tid * KT_PITCH];
#pragma unroll
    for (int dd = 0; dd < D_CHUNK; dd += 4) {
      const float4 kv = *(const float4*)&ktr[dd];
      const float4 wv = *(const float4*)&wsh[d0 + dd];
#pragma unroll
      for (int u = 0; u < 4; ++u) {
        const float kvu = fget(kv, u);
        const float wvu = fget(wv, u);
        const float4 qv = *(const float4*)&qsh4[(d0 + dd + u) * Q_TILE]; // broadcast
        acc[0] = fmaf(fast_tanh(qv.x + kvu), wvu, acc[0]);
        acc[1] = fmaf(fast_tanh(qv.y + kvu), wvu, acc[1]);
        acc[2] = fmaf(fast_tanh(qv.z + kvu), wvu, acc[2]);
        acc[3] = fmaf(fast_tanh(qv.w + kvu), wvu, acc[3]);
      }
    }
  }

  const float bl = b_logit[0];
  const size_t obase = (size_t)(b * LQ + q0) * LKV + k;
#pragma unroll
  for (int qi = 0; qi < Q_TILE; ++qi)
    out[obase + (size_t)qi * LKV] = acc[qi] + bl;   // coalesced over tid
}

// =====================================================================
// Kernel 3: in-place masked softmax over k for each (b,q) row of d_out.
//   inv = !mask; if ALL inv in a row -> no masking (reference rule).
// =====================================================================
__global__ __launch_bounds__(256) void mask_softmax_kernel(
    float* __restrict__ out,                 // B x LQ x LKV (logits -> weights)
    const unsigned char* __restrict__ mask)  // B x LQ x LKV (bool)
{
  __shared__ float red[8];
  __shared__ int   wany[8];
  __shared__ int   s_any;
  __shared__ float s_bcast;

  const int tid  = threadIdx.x;
  const int lane = tid & 31;
  const int wave = tid >> 5;
  const size_t row = (size_t)blockIdx.x * LKV;

  float l0 = out[row + tid];
  float l1 = out[row + tid + 256];
  const unsigned char m0 = mask[row + tid];
  const unsigned char m1 = mask[row + tid + 256];

  const unsigned long long bal = __ballot(m0 | m1);
  if (lane == 0) wany[wave] = (bal != 0ull) ? 1 : 0;
  __syncthreads();
  if (tid == 0) {
    int a = 0;
#pragma unroll
    for (int i = 0; i < 8; ++i) a |= wany[i];
    s_any = a;
  }
  __syncthreads();

  if (s_any) {
    if (!m0) l0 = NEG_INF;
    if (!m1) l1 = NEG_INF;
  }

  // block max
  float mx = fmaxf(l0, l1);
#pragma unroll
  for (int off = 16; off > 0; off >>= 1)
    mx = fmaxf(mx, __shfl_xor(mx, off, 32));
  if (lane == 0) red[wave] = mx;
  __syncthreads();
  if (tid == 0) {
    float m = red[0];
#pragma unroll
    for (int i = 1; i < 8; ++i) m = fmaxf(m, red[i]);
    s_bcast = m;
  }
  __syncthreads();
  const float rowmax = s_bcast;

  const float e0 = __expf(l0 - rowmax);
  const float e1 = __expf(l1 - rowmax);
  float s = e0 + e1;
#pragma unroll
  for (int off = 16; off > 0; off >>= 1)
    s += __shfl_xor(s, off, 32);
  __syncthreads();                 // red[] reuse: prior readers done
  if (lane == 0) red[wave] = s;
  __syncthreads();
  if (tid == 0) {
    float t = 0.0f;
#pragma unroll
    for (int i = 0; i < 8; ++i) t += red[i];
    s_bcast = t;
  }
  __syncthreads();
  const float inv = 1.0f / s_bcast;

  out[row + tid]       = e0 * inv;
  out[row + tid + 256] = e1 * inv;
}

// =====================================================================
extern "C" void kernel_launch(void* const* d_in, const int* in_sizes, int n_in,
                              void* d_out, int out_size, void* d_ws, size_t ws_size,
                              hipStream_t stream) {
  const float*         queries  = (const float*)d_in[0];         // 4*256*256
  const float*         keys     = (const float*)d_in[1];         // 4*512*256
  // d_in[2] = values: unused by reference output
  const unsigned char* mask     = (const unsigned char*)d_in[3]; // 4*256*512 bool
  const float*         W_concat = (const float*)d_in[4];         // 256 x 512
  const float*         b_concat = (const float*)d_in[5];         // 256
  const float*         W_logit  = (const float*)d_in[6];         // 256
  const float*         b_logit  = (const float*)d_in[7];         // 1
  float*               out      = (float*)d_out;                 // 4*256*512

  float* qp = (float*)d_ws;                 // 1024 x 256 = 1 MB
  float* kp = qp + (size_t)1024 * D_DIM;    // 2048 x 256 = 2 MB

  // qp = Q @ Wq^T : 1024 tiles, 8 waves/block -> 128 blocks
  proj_wmma_kernel<<<128, 256, 0, stream>>>(queries, W_concat, nullptr, qp, 1024);
  // kp = K @ Wk^T + b : 2048 tiles -> 256 blocks; Wk = W_concat[:, 256:]
  proj_wmma_kernel<<<256, 256, 0, stream>>>(keys, W_concat + 256, b_concat, kp, 2048);
  // raw logits into d_out: 512 blocks
  score_kernel<<<BATCH * (LQ / Q_TILE) * (LKV / K_BLK), 256, 0, stream>>>(
      qp, kp, W_logit, b_logit, out);
  // in-place masked softmax per (b,q) row
  mask_softmax_kernel<<<BATCH * LQ, 256, 0, stream>>>(out, mask);
}